// QCNet_65481071405001
// MI455X (gfx1250) — compile-verified
//
#include <hip/hip_runtime.h>
#include <math.h>
#include <stdint.h>

// ---------------------------------------------------------------------------
// QCNet forward for MI455X (gfx1250, wave32).
//  - fc1 GEMM: V_WMMA_F32_16X16X4_F32 (exact fp32, matches reference)
//  - A-operand staged to LDS by the Tensor Data Mover (tensor_load_to_lds,
//    TENSORcnt double-buffered). This toolchain's builtin takes 6 args.
// ---------------------------------------------------------------------------

typedef __attribute__((ext_vector_type(2))) float v2f;
typedef __attribute__((ext_vector_type(8))) float v8f;
typedef __attribute__((ext_vector_type(4))) unsigned int u32x4;
typedef __attribute__((ext_vector_type(8))) int i32x8;
typedef __attribute__((ext_vector_type(4))) int i32x4;

#define B       128
#define C1      6
#define C2      15
#define H0      224
#define OH1     111   // conv1 out
#define PH1     110   // pool1 out
#define OH2     55    // conv2 out
#define PH2     54    // pool2 out
#define SPAT2   (PH2*PH2)          // 2916
#define FLAT    (C2*SPAT2)         // 43740
#define FC1_N   120
#define FC2_N   84
#define KCC     324                // K sub-chunk (2916 = 9*324); 16B-aligned rows
#define NSUB    9                  // sub-chunks per channel
#define NCHUNK  (C2*NSUB)          // 135

#ifndef __has_builtin
#define __has_builtin(x) 0
#endif

#if __has_builtin(__builtin_amdgcn_tensor_load_to_lds) && \
    __has_builtin(__builtin_amdgcn_s_wait_tensorcnt)
#define USE_TDM 1
#else
#define USE_TDM 0
#warning "fc1_wmma: tensor_load_to_lds builtin unavailable -> sync LDS staging fallback"
#endif

// ---------------- conv1: 3->6, 5x5, stride 2, pad 1, + bias + relu ----------
__global__ void conv1_relu(const float* __restrict__ x,
                           const float* __restrict__ w,
                           const float* __restrict__ bias,
                           float* __restrict__ out) {
    int idx = blockIdx.x * 256 + threadIdx.x;
    const int total = B * C1 * OH1 * OH1;
    if (idx >= total) return;
    int ow = idx % OH1; int t = idx / OH1;
    int oh = t % OH1;   t /= OH1;
    int oc = t % C1;    int b = t / C1;

    float acc = bias[oc];
    int ih0 = oh * 2 - 1, iw0 = ow * 2 - 1;
    const float* xb = x + (size_t)b * 3 * H0 * H0;
    const float* wk = w + oc * 75;
    #pragma unroll
    for (int ic = 0; ic < 3; ++ic) {
        #pragma unroll
        for (int kh = 0; kh < 5; ++kh) {
            int ih = ih0 + kh;
            if (ih < 0 || ih >= H0) continue;
            const float* xr = xb + ((size_t)ic * H0 + ih) * H0;
            #pragma unroll
            for (int kw = 0; kw < 5; ++kw) {
                int iw = iw0 + kw;
                if (iw < 0 || iw >= H0) continue;
                acc = fmaf(xr[iw], wk[(ic * 5 + kh) * 5 + kw], acc);
            }
        }
    }
    out[idx] = fmaxf(acc, 0.0f);
}

// ---------------- 2x2 stride-1 max pool (generic C,H) -----------------------
__global__ void maxpool2_s1(const float* __restrict__ in, float* __restrict__ out,
                            int C, int H) {
    int OH = H - 1;
    int total = B * C * OH * OH;
    int idx = blockIdx.x * 256 + threadIdx.x;
    if (idx >= total) return;
    int ow = idx % OH; int t = idx / OH;
    int oh = t % OH;   t /= OH;
    int c  = t % C;    int b = t / C;
    const float* p = in + (((size_t)b * C + c) * H + oh) * H + ow;
    float m = fmaxf(fmaxf(p[0], p[1]), fmaxf(p[H], p[H + 1]));
    out[idx] = m;
}

// ---------------- conv2: 6->15, 3x3, stride 2, pad 1, + bias + relu ---------
__global__ void conv2_relu(const float* __restrict__ in,
                           const float* __restrict__ w,
                           const float* __restrict__ bias,
                           float* __restrict__ out) {
    int idx = blockIdx.x * 256 + threadIdx.x;
    const int total = B * C2 * OH2 * OH2;
    if (idx >= total) return;
    int ow = idx % OH2; int t = idx / OH2;
    int oh = t % OH2;   t /= OH2;
    int oc = t % C2;    int b = t / C2;

    float acc = bias[oc];
    int ih0 = oh * 2 - 1, iw0 = ow * 2 - 1;
    const float* xb = in + (size_t)b * C1 * PH1 * PH1;
    const float* wk = w + oc * (C1 * 9);
    #pragma unroll
    for (int ic = 0; ic < C1; ++ic) {
        #pragma unroll
        for (int kh = 0; kh < 3; ++kh) {
            int ih = ih0 + kh;
            if (ih < 0 || ih >= PH1) continue;
            const float* xr = xb + ((size_t)ic * PH1 + ih) * PH1;
            #pragma unroll
            for (int kw = 0; kw < 3; ++kw) {
                int iw = iw0 + kw;
                if (iw < 0 || iw >= PH1) continue;
                acc = fmaf(xr[iw], wk[(ic * 3 + kh) * 3 + kw], acc);
            }
        }
    }
    out[idx] = fmaxf(acc, 0.0f);
}

// ---------------- SE squeeze: mean over 54x54 per (b,c) ---------------------
__global__ void se_mean(const float* __restrict__ pool2, float* __restrict__ s) {
    int bc = blockIdx.x;                 // b*15 + c
    const float* p = pool2 + (size_t)bc * SPAT2;
    float acc = 0.0f;
    for (int i = threadIdx.x; i < SPAT2; i += 256) acc += p[i];
    __shared__ float red[256];
    red[threadIdx.x] = acc;
    __syncthreads();
    for (int off = 128; off > 0; off >>= 1) {
        if (threadIdx.x < off) red[threadIdx.x] += red[threadIdx.x + off];
        __syncthreads();
    }
    if (threadIdx.x == 0) s[bc] = red[0] * (1.0f / (float)SPAT2);
}

// ---------------- SE excitation: sigmoid(relu(s@w1)*w2) ---------------------
__global__ void se_att(const float* __restrict__ s,
                       const float* __restrict__ w1,   // [15]  (att_w1: 15->1)
                       const float* __restrict__ w2,   // [15]  (att_w2: 1->15)
                       float* __restrict__ att) {
    int b = blockIdx.x;
    __shared__ float t;
    if (threadIdx.x == 0) {
        float a = 0.0f;
        for (int c = 0; c < C2; ++c) a = fmaf(s[b * C2 + c], w1[c], a);
        t = fmaxf(a, 0.0f);
    }
    __syncthreads();
    if (threadIdx.x < C2) {
        float z = t * w2[threadIdx.x];
        att[b * C2 + threadIdx.x] = 1.0f / (1.0f + expf(-z));
    }
}

#if USE_TDM
// Issue one TDM 2D-tile load: 16 rows x KCC fp32, row stride FLAT elements,
// global -> LDS (contiguous, row-major, stride KCC). D# per ISA ch8.
__device__ __forceinline__ void tdm_issue_tile(const float* gsrc, float* ldsdst) {
    unsigned lds_off = (unsigned)(uintptr_t)ldsdst;  // generic addr[31:0] == LDS byte offset
    unsigned long long ga = (unsigned long long)(uintptr_t)gsrc;

    u32x4 g0;
    g0[0] = 1u;                                            // count=1 (valid), user mode
    g0[1] = lds_off;                                       // lds_addr [63:32]
    g0[2] = (unsigned)ga;                                  // global_addr [95:64]
    g0[3] = ((unsigned)(ga >> 32) & 0x01FFFFFFu)           // global_addr [120:96]
            | 0x80000000u;                                 // type=2 ("image") [127:126]

    i32x8 g1;
    g1[0] = 0x00020000;                                    // data_size=2 (4B), no multicast
    g1[1] = (int)((SPAT2 & 0xFFFFu) << 16);                // tensor_dim0[15:0]
    g1[2] = (int)((SPAT2 >> 16) | (16u << 16));            // tensor_dim0[31:16] | tensor_dim1[15:0]
    g1[3] = (int)((unsigned)KCC << 16);                    // tensor_dim1[31:16]=0 | tile_dim0
    g1[4] = 16;                                            // tile_dim1=16, tile_dim2=0
    g1[5] = (int)FLAT;                                     // tensor_dim0_stride[31:0]=43740
    g1[6] = 0;                                             // stride0[47:32]=0, stride1[15:0]=0
    g1[7] = 0;
    i32x4 z4 = {0, 0, 0, 0};                               // groups 2/3 unused (2D)
    i32x8 z8 = {0, 0, 0, 0, 0, 0, 0, 0};
    // 6-arg toolchain variant: (g0, g1, g2, g3, extra, cpol)
    __builtin_amdgcn_tensor_load_to_lds(g0, g1, z4, z4, z8, 0);
}
#endif

// ---------------- fc1 via V_WMMA_F32_16X16X4_F32 ----------------------------
// out[m,n] = relu( sum_c att[m,c] * sum_s pool2[m,c,s]*W[n,c*2916+s] + b[n] )
// grid = 8 blocks (16-row M tiles), 256 threads = 8 waves (16-col N tiles).
__global__ __launch_bounds__(256) void fc1_wmma(
        const float* __restrict__ pool2,
        const float* __restrict__ att,
        const float* __restrict__ fc1w,   // [120][43740]
        const float* __restrict__ fc1b,
        float* __restrict__ f1) {         // [128][120]
    __shared__ float Alds[2][16 * KCC];   // 2 x 20.25 KB double buffer
    __shared__ float att_lds[16 * C2];

    const int mbase = blockIdx.x * 16;
    const int wave  = threadIdx.x >> 5;   // N tile id 0..7
    const int lane  = threadIdx.x & 31;
    const int half  = lane >> 4;          // 0: K+0/+1   1: K+2/+3
    const int l15   = lane & 15;
    const int n     = wave * 16 + l15;    // output column
    const bool nvalid = (n < FC1_N);
    const float* wrow = fc1w + (size_t)(nvalid ? n : 0) * FLAT;
    const float* abase = pool2 + (size_t)mbase * FLAT;

    // stage SE attention scales for this M tile
    for (int idx = threadIdx.x; idx < 16 * C2; idx += 256) {
        int r = idx / C2, c = idx - r * C2;
        att_lds[idx] = att[(mbase + r) * C2 + c];
    }

    v8f acc  = {};
    v8f accC = {};

#if USE_TDM
    if (wave == 0) tdm_issue_tile(abase, &Alds[0][0]);     // prologue: chunk 0
#endif

    for (int q = 0; q < NCHUNK; ++q) {
        const int c     = q / NSUB;
        const int s2    = q - c * NSUB;
        const int kbase = c * SPAT2 + s2 * KCC;
        const int cur   = q & 1;

#if USE_TDM
        if (q + 1 < NCHUNK) {
            const int c1 = (q + 1) / NSUB;
            const int s1 = (q + 1) - c1 * NSUB;
            if (wave == 0) {
                tdm_issue_tile(abase + c1 * SPAT2 + s1 * KCC, &Alds[(q + 1) & 1][0]);
                __builtin_amdgcn_s_wait_tensorcnt(1);      // chunk q done, q+1 in flight
            }
        } else if (wave == 0) {
            __builtin_amdgcn_s_wait_tensorcnt(0);
        }
        __syncthreads();                                   // publish DMA'd tile
#else
        __syncthreads();                                   // WAR on LDS buffer
        for (int idx = threadIdx.x; idx < 16 * KCC; idx += 256) {
            int r = idx / KCC, kk = idx - r * KCC;
            Alds[cur][idx] = abase[(size_t)r * FLAT + kbase + kk];
        }
        __syncthreads();
#endif

        if (nvalid) __builtin_prefetch(wrow + kbase + KCC, 0, 0);  // global_prefetch_b8

        const float* Ab = &Alds[cur][l15 * KCC + half * 2];
        for (int i = 0; i < KCC; i += 4) {
            v2f a;
            a.x = Ab[i];
            a.y = Ab[i + 1];
            v2f bfrag = {0.0f, 0.0f};
            if (nvalid) bfrag = *(const v2f*)(wrow + kbase + i + half * 2);  // 8B aligned
            accC = __builtin_amdgcn_wmma_f32_16x16x4_f32(
                false, a, false, bfrag, (short)0, accC, false, false);
        }

        if (s2 == NSUB - 1) {            // channel boundary: fold in SE scale
            #pragma unroll
            for (int r = 0; r < 8; ++r) {
                acc[r] += att_lds[(r + half * 8) * C2 + c] * accC[r];
                accC[r] = 0.0f;
            }
        }
#if USE_TDM
        __syncthreads();                 // all readers done before DMA reuses buffer
#endif
    }

    // D layout: vgpr r -> row (r + 8*half), col = l15
    if (nvalid) {
        float bia = fc1b[n];
        #pragma unroll
        for (int r = 0; r < 8; ++r) {
            int m = mbase + r + half * 8;
            f1[m * FC1_N + n] = fmaxf(acc[r] + bia, 0.0f);
        }
    }
}

// ---------------- fc2 + fc3 + quantum head ----------------------------------
__global__ void head_kernel(const float* __restrict__ f1,
                            const float* __restrict__ fc2w, const float* __restrict__ fc2b,
                            const float* __restrict__ fc3w, const float* __restrict__ fc3b,
                            float* __restrict__ out) {
    int b = blockIdx.x;
    __shared__ float f2[FC2_N];
    int j = threadIdx.x;
    if (j < FC2_N) {
        float a = fc2b[j];
        const float* wr = fc2w + j * FC1_N;
        const float* fr = f1 + b * FC1_N;
        for (int i = 0; i < FC1_N; ++i) a = fmaf(fr[i], wr[i], a);
        f2[j] = fmaxf(a, 0.0f);
    }
    __syncthreads();
    if (j == 0) {
        float l = fc3b[0];
        for (int i = 0; i < FC2_N; ++i) l = fmaf(f2[i], fc3w[i], l);
        float p = 1.0f / (1.0f + expf(-cosf(l)));
        out[b * 2 + 0] = p;
        out[b * 2 + 1] = 1.0f - p;
    }
}

// ---------------------------------------------------------------------------
extern "C" void kernel_launch(void* const* d_in, const int* in_sizes, int n_in,
                              void* d_out, int out_size, void* d_ws, size_t ws_size,
                              hipStream_t stream) {
    const float* x       = (const float*)d_in[0];
    const float* conv1_w = (const float*)d_in[1];
    const float* conv1_b = (const float*)d_in[2];
    const float* conv2_w = (const float*)d_in[3];
    const float* conv2_b = (const float*)d_in[4];
    const float* att_w1  = (const float*)d_in[5];
    const float* att_w2  = (const float*)d_in[6];
    const float* fc1_w   = (const float*)d_in[7];
    const float* fc1_b   = (const float*)d_in[8];
    const float* fc2_w   = (const float*)d_in[9];
    const float* fc2_b   = (const float*)d_in[10];
    const float* fc3_w   = (const float*)d_in[11];
    const float* fc3_b   = (const float*)d_in[12];
    float* out = (float*)d_out;

    // workspace layout (floats); regions aliased over the pipeline (~75 MB peak)
    float* ws = (float*)d_ws;
    const size_t SZ_A = (size_t)B * C1 * OH1 * OH1;   // conv1 out, then conv2 out
    const size_t SZ_P = (size_t)B * C1 * PH1 * PH1;   // pool1 out, then pool2 out
    float* bufA = ws;
    float* bufP = ws + SZ_A;
    float* sbuf = ws + SZ_A + SZ_P;      // [128*15] squeeze means
    float* attb = sbuf + B * C2;         // [128*15] attention scales
    float* f1   = attb + B * C2;         // [128*120]

    const int tconv1 = B * C1 * OH1 * OH1;
    const int tpool1 = B * C1 * PH1 * PH1;
    const int tconv2 = B * C2 * OH2 * OH2;
    const int tpool2 = B * C2 * PH2 * PH2;

    conv1_relu<<<(tconv1 + 255) / 256, 256, 0, stream>>>(x, conv1_w, conv1_b, bufA);
    maxpool2_s1<<<(tpool1 + 255) / 256, 256, 0, stream>>>(bufA, bufP, C1, OH1);
    conv2_relu<<<(tconv2 + 255) / 256, 256, 0, stream>>>(bufP, conv2_w, conv2_b, bufA);
    maxpool2_s1<<<(tpool2 + 255) / 256, 256, 0, stream>>>(bufA, bufP, C2, OH2);
    se_mean<<<B * C2, 256, 0, stream>>>(bufP, sbuf);
    se_att<<<B, 32, 0, stream>>>(sbuf, att_w1, att_w2, attb);
    fc1_wmma<<<8, 256, 0, stream>>>(bufP, attb, fc1_w, fc1_b, f1);
    head_kernel<<<B, 128, 0, stream>>>(f1, fc2_w, fc2_b, fc3_w, fc3_b, out);
}